// RobustInverseCompositional_28484223107707
// MI455X (gfx1250) — compile-verified
//
#include <hip/hip_runtime.h>
#include <math.h>

// ---------------- problem constants ----------------
#define BB   16
#define HH   512
#define WW   512
#define ROWS 8                 // rows per block
#define NBLK (HH / ROWS)       // 64 blocks per batch
#define LAM2 0.01f
#define TOL  0.001f
#define NITER 10

// ws layout (floats): [0..31] p, [32] err, [64 ..] partials P[b][r][5]
#define WS_P    0
#define WS_ERR  32
#define WS_PART 64

typedef float v2f __attribute__((ext_vector_type(2)));
typedef float v8f __attribute__((ext_vector_type(8)));

__device__ __forceinline__ float cubic_w(float x) {
    float ax = fabsf(x);
    float ax2 = ax * ax;
    float ax3 = ax2 * ax;
    if (ax <= 1.0f) return 1.5f * ax3 - 2.5f * ax2 + 1.0f;
    if (ax <  2.0f) return -0.5f * ax3 + 2.5f * ax2 - 4.0f * ax + 2.0f;
    return 0.0f;
}

__device__ __forceinline__ int clampi(int v, int lo, int hi) {
    return min(max(v, lo), hi);
}

// ---------------- init: p -> ws, err = 1e10 ----------------
__global__ void ric_init(const float* __restrict__ p, float* __restrict__ ws) {
    int t = threadIdx.x;
    if (t < 32) ws[WS_P + t] = p[t];
    if (t == 32) ws[WS_ERR] = 1e10f;
}

// ---------------- fused warp + gradient + robust reduction ----------------
// grid: (NBLK, BB), block: 256. Produces P[b][blk][5] partials.
__global__ void ric_reduce(const float* __restrict__ I1,
                           const float* __restrict__ I2,
                           float* __restrict__ ws) {
    __shared__ float sI2[11 * WW];   // rows y0+ity-1 .. y0+ity+9 (clamped)
    __shared__ float sI1[10 * WW];   // rows y0-1 .. y0+8 (clamped)
    __shared__ float sred[5 * 256];

    const int t   = threadIdx.x;
    const int b   = blockIdx.y;
    const int blk = blockIdx.x;
    const int y0  = blk * ROWS;

    // per-batch warp parameters (constant fractional offsets!)
    const float tx = ws[WS_P + 2 * b];
    const float ty = ws[WS_P + 2 * b + 1];
    const int   itx = (int)floorf(tx);
    const int   ity = (int)floorf(ty);
    const float dx = tx - (float)itx;
    const float dy = ty - (float)ity;
    const float wx0 = cubic_w(dx + 1.0f), wx1 = cubic_w(dx);
    const float wx2 = cubic_w(dx - 1.0f), wx3 = cubic_w(dx - 2.0f);
    const float wy0 = cubic_w(dy + 1.0f), wy1 = cubic_w(dy);
    const float wy2 = cubic_w(dy - 1.0f), wy3 = cubic_w(dy - 2.0f);

    // stage 11 rows of I2 (clamped), vectorized 128 float4 per row
    {
        const float4* I2v = (const float4*)I2;
        for (int idx = t; idx < 11 * 128; idx += 256) {
            int k  = idx >> 7;
            int c4 = idx & 127;
            int gy = clampi(y0 + ity - 1 + k, 0, HH - 1);
            ((float4*)sI2)[k * 128 + c4] = I2v[((size_t)b * HH + gy) * 128 + c4];
        }
        const float4* I1v = (const float4*)I1;
        for (int idx = t; idx < 10 * 128; idx += 256) {
            int k  = idx >> 7;
            int c4 = idx & 127;
            int gy = clampi(y0 - 1 + k, 0, HH - 1);
            ((float4*)sI1)[k * 128 + c4] = I1v[((size_t)b * HH + gy) * 128 + c4];
        }
    }
    __syncthreads();

    float hxx = 0.f, hxy = 0.f, hyy = 0.f, bx = 0.f, by = 0.f;

    #pragma unroll 4
    for (int k = 0; k < 16; ++k) {
        int pi = (k << 8) + t;          // 0..4095
        int ly = pi >> 9;               // local row 0..7
        int x  = pi & (WW - 1);
        int y  = y0 + ly;

        int xb = x + itx - 1;
        int xi0 = clampi(xb + 0, 0, WW - 1);
        int xi1 = clampi(xb + 1, 0, WW - 1);
        int xi2 = clampi(xb + 2, 0, WW - 1);
        int xi3 = clampi(xb + 3, 0, WW - 1);

        const float* r0 = &sI2[(ly + 0) * WW];
        const float* r1 = &sI2[(ly + 1) * WW];
        const float* r2 = &sI2[(ly + 2) * WW];
        const float* r3 = &sI2[(ly + 3) * WW];
        float s0 = wx0 * r0[xi0] + wx1 * r0[xi1] + wx2 * r0[xi2] + wx3 * r0[xi3];
        float s1 = wx0 * r1[xi0] + wx1 * r1[xi1] + wx2 * r1[xi2] + wx3 * r1[xi3];
        float s2 = wx0 * r2[xi0] + wx1 * r2[xi1] + wx2 * r2[xi2] + wx3 * r2[xi3];
        float s3 = wx0 * r3[xi0] + wx1 * r3[xi1] + wx2 * r3[xi2] + wx3 * r3[xi3];
        float Iw = wy0 * s0 + wy1 * s1 + wy2 * s2 + wy3 * s3;

        const float* c1 = &sI1[(ly + 1) * WW];
        float c   = c1[x];
        float Ixg = (x > 0 && x < WW - 1) ? 0.5f * (c1[x + 1] - c1[x - 1]) : 0.0f;
        float Iyg = (y > 0 && y < HH - 1)
                        ? 0.5f * (sI1[(ly + 2) * WW + x] - sI1[ly * WW + x])
                        : 0.0f;

        float DI  = Iw - c;
        float rho = __frsqrt_rn(DI * DI + LAM2);
        hxx += rho * Ixg * Ixg;
        hxy += rho * Ixg * Iyg;
        hyy += rho * Iyg * Iyg;
        bx  += rho * DI * Ixg;
        by  += rho * DI * Iyg;
    }

    // deterministic LDS tree reduction of 5 quantities
    sred[0 * 256 + t] = hxx;
    sred[1 * 256 + t] = hxy;
    sred[2 * 256 + t] = hyy;
    sred[3 * 256 + t] = bx;
    sred[4 * 256 + t] = by;
    __syncthreads();
    for (int off = 128; off > 0; off >>= 1) {
        if (t < off) {
            #pragma unroll
            for (int q = 0; q < 5; ++q)
                sred[q * 256 + t] += sred[q * 256 + t + off];
        }
        __syncthreads();
    }
    if (t == 0) {
        float* P = ws + WS_PART + ((size_t)b * NBLK + blk) * 5;
        #pragma unroll
        for (int q = 0; q < 5; ++q) P[q] = sred[q * 256];
    }
}

// ---------------- solve: WMMA reduction of partials + 2x2 GN update ----------------
// single wave32. Batches occupy the N=16 dimension of V_WMMA_F32_16X16X4_F32;
// A = ones(16x4) so D[0][n] accumulates column sums: 16 WMMAs reduce 64 partials.
__global__ void ric_solve(float* __restrict__ ws) {
    const int L    = threadIdx.x;     // 0..31
    const int bb   = L & 15;          // batch = N column
    const int half = L >> 4;          // which K slots this lane feeds
    const float* P = ws + WS_PART;

    v2f onesA;
    onesA.x = 1.0f; onesA.y = 1.0f;

    float S[5];
    #pragma unroll
    for (int q = 0; q < 5; ++q) {
        v8f acc = {};
        for (int tstep = 0; tstep < 16; ++tstep) {
            int r0 = 4 * tstep + half;        // lanes 0-15: rows 4t,   lanes 16-31: 4t+1
            int r1 = 4 * tstep + 2 + half;    // lanes 0-15: rows 4t+2, lanes 16-31: 4t+3
            v2f bm;
            bm.x = P[((size_t)bb * NBLK + r0) * 5 + q];
            bm.y = P[((size_t)bb * NBLK + r1) * 5 + q];
            acc = __builtin_amdgcn_wmma_f32_16x16x4_f32(
                false, onesA, false, bm, (short)0, acc, false, false);
        }
        S[q] = acc[0];   // D[0][bb] (rows identical since A is all-ones)
    }

    float Hxx = S[0], Hxy = S[1], Hyy = S[2], Bx = S[3], By = S[4];
    float det = Hxx * Hyy - Hxy * Hxy;
    float inv = 1.0f / det;
    float dpx = inv * ( Hyy * Bx - Hxy * By);
    float dpy = inv * (-Hxy * Bx + Hxx * By);

    __shared__ float sq[16];
    if (half == 0) sq[bb] = dpx * dpx + dpy * dpy;
    __syncthreads();

    float errold = ws[WS_ERR];
    bool active  = errold > TOL;
    if (L == 0) {
        float s = 0.f;
        #pragma unroll
        for (int i = 0; i < 16; ++i) s += sq[i];
        ws[WS_ERR] = active ? sqrtf(s) : errold;
    }
    if (half == 0 && active) {
        ws[WS_P + 2 * bb]     -= dpx;
        ws[WS_P + 2 * bb + 1] -= dpy;
    }
}

// ---------------- final warp: write p, err, DI, Iw ----------------
__global__ void ric_final(const float* __restrict__ I1,
                          const float* __restrict__ I2,
                          const float* __restrict__ ws,
                          float* __restrict__ out) {
    __shared__ float sI2[11 * WW];

    const int t   = threadIdx.x;
    const int b   = blockIdx.y;
    const int blk = blockIdx.x;
    const int y0  = blk * ROWS;

    if (b == 0 && blk == 0) {
        if (t < 32)  out[t]  = ws[WS_P + t];
        if (t == 32) out[32] = ws[WS_ERR];
    }

    const float tx = ws[WS_P + 2 * b];
    const float ty = ws[WS_P + 2 * b + 1];
    const int   itx = (int)floorf(tx);
    const int   ity = (int)floorf(ty);
    const float dx = tx - (float)itx;
    const float dy = ty - (float)ity;
    const float wx0 = cubic_w(dx + 1.0f), wx1 = cubic_w(dx);
    const float wx2 = cubic_w(dx - 1.0f), wx3 = cubic_w(dx - 2.0f);
    const float wy0 = cubic_w(dy + 1.0f), wy1 = cubic_w(dy);
    const float wy2 = cubic_w(dy - 1.0f), wy3 = cubic_w(dy - 2.0f);

    {
        const float4* I2v = (const float4*)I2;
        for (int idx = t; idx < 11 * 128; idx += 256) {
            int k  = idx >> 7;
            int c4 = idx & 127;
            int gy = clampi(y0 + ity - 1 + k, 0, HH - 1);
            ((float4*)sI2)[k * 128 + c4] = I2v[((size_t)b * HH + gy) * 128 + c4];
        }
    }
    __syncthreads();

    const size_t DI_OFF = 33;
    const size_t IW_OFF = 33 + (size_t)BB * HH * WW;

    #pragma unroll 4
    for (int k = 0; k < 16; ++k) {
        int pi = (k << 8) + t;
        int ly = pi >> 9;
        int x  = pi & (WW - 1);
        int y  = y0 + ly;

        int xb = x + itx - 1;
        int xi0 = clampi(xb + 0, 0, WW - 1);
        int xi1 = clampi(xb + 1, 0, WW - 1);
        int xi2 = clampi(xb + 2, 0, WW - 1);
        int xi3 = clampi(xb + 3, 0, WW - 1);

        const float* r0 = &sI2[(ly + 0) * WW];
        const float* r1 = &sI2[(ly + 1) * WW];
        const float* r2 = &sI2[(ly + 2) * WW];
        const float* r3 = &sI2[(ly + 3) * WW];
        float s0 = wx0 * r0[xi0] + wx1 * r0[xi1] + wx2 * r0[xi2] + wx3 * r0[xi3];
        float s1 = wx0 * r1[xi0] + wx1 * r1[xi1] + wx2 * r1[xi2] + wx3 * r1[xi3];
        float s2 = wx0 * r2[xi0] + wx1 * r2[xi1] + wx2 * r2[xi2] + wx3 * r2[xi3];
        float s3 = wx0 * r3[xi0] + wx1 * r3[xi1] + wx2 * r3[xi2] + wx3 * r3[xi3];
        float Iw = wy0 * s0 + wy1 * s1 + wy2 * s2 + wy3 * s3;

        size_t pix = ((size_t)b * HH + y) * WW + x;
        float  c   = I1[pix];
        out[DI_OFF + pix] = Iw - c;
        out[IW_OFF + pix] = Iw;
    }
}

// ---------------- host entry ----------------
extern "C" void kernel_launch(void* const* d_in, const int* in_sizes, int n_in,
                              void* d_out, int out_size, void* d_ws, size_t ws_size,
                              hipStream_t stream) {
    const float* I1 = (const float*)d_in[0];
    const float* I2 = (const float*)d_in[1];
    const float* p  = (const float*)d_in[2];
    float* out = (float*)d_out;
    float* ws  = (float*)d_ws;

    ric_init<<<1, 64, 0, stream>>>(p, ws);
    for (int it = 0; it < NITER; ++it) {
        ric_reduce<<<dim3(NBLK, BB), 256, 0, stream>>>(I1, I2, ws);
        ric_solve<<<1, 32, 0, stream>>>(ws);
    }
    ric_final<<<dim3(NBLK, BB), 256, 0, stream>>>(I1, I2, ws, out);
}